// ShortConv1d_14456859918520
// MI455X (gfx1250) — compile-verified
//
#include <hip/hip_runtime.h>

// Depthwise causal conv1d: y[b,t,d] = sum_{k<4} w[d,k] * x[b, t-3+k, d]
// Shapes fixed by the reference: x (4, 4096, 2048) f32, w (2048, 4) f32.
//
// Memory-bound (1 FLOP/byte, ~268 MB moved -> ~12 us floor at 23.3 TB/s).
// CDNA5 path: async global->LDS b128 staging (ASYNCcnt) with pipelined
// s_wait_asynccnt drains, register sliding window, NT stores for y.
// No WMMA: depthwise conv has no channel contraction, so matrix ops cannot
// reduce data movement for this op.

namespace {
constexpr int kB   = 4;
constexpr int kT   = 4096;
constexpr int kD   = 2048;
constexpr int kDV  = kD / 4;   // float4 columns per (b,t) row = 512
constexpr int kBLK = 128;      // threads per block == columns per block
constexpr int kTT  = 16;       // time rows staged per block (32 KB LDS)
}

typedef float v4f __attribute__((ext_vector_type(4)));

__global__ __launch_bounds__(kBLK)
void dwconv1d_async_kernel(const float* __restrict__ x,
                           const float* __restrict__ w,
                           float* __restrict__ y) {
    __shared__ float4 tile[kTT * kBLK];   // 16 rows x 128 cols x 16 B = 32 KB

    const int tid  = threadIdx.x;
    const int dvec = blockIdx.x * kBLK + tid;   // which float4 column of D
    const int t0   = blockIdx.y * kTT;          // first time row of this chunk
    const int b    = blockIdx.z;

    const float4* xv = (const float4*)x;
    float4*       yv = (float4*)y;

    // flat float4 index of (b, t, dvec) is col + t*kDV
    const size_t col = (size_t)b * kT * kDV + (size_t)dvec;

    // ---- issue async global->LDS b128 loads for rows t0 .. t0+kTT-1 ----
    // LDS byte offset of this thread's column slot (low 32 bits of the
    // generic address of an LDS object are the LDS byte offset).
    const unsigned ldsColBase = (unsigned)(size_t)(void*)&tile[tid];
    // uniform SGPR base = &x[b, t0, 0]; per-lane VGPR byte offset = dvec*16;
    // per-row stride via the signed 24-bit instruction offset (r * 8192 B).
    const char* sbase = (const char*)x + ((size_t)b * kT + (size_t)t0) * kD * 4;
    const unsigned voff = (unsigned)dvec * 16u;
#pragma unroll
    for (int r = 0; r < kTT; ++r) {
        asm volatile("global_load_async_to_lds_b128 %0, %1, %2 offset:%3"
                     :
                     : "v"(ldsColBase + (unsigned)(r * kBLK * 16)),
                       "v"(voff),
                       "s"(sbase),
                       "i"(r * kD * 4)
                     : "memory");
    }

    // ---- causal halo rows t0-3..t0-1 (zeros for the first chunk);
    //      latency overlaps the async stream ----
    float4 xm3 = make_float4(0.f, 0.f, 0.f, 0.f);
    float4 xm2 = xm3;
    float4 xm1 = xm3;
    if (t0 > 0) {   // kTT >= 3, so t0 > 0 implies all three rows exist
        xm3 = xv[col + (size_t)(t0 - 3) * kDV];
        xm2 = xv[col + (size_t)(t0 - 2) * kDV];
        xm1 = xv[col + (size_t)(t0 - 1) * kDV];
    }

    // ---- per-channel taps for channels 4*dvec .. 4*dvec+3 ----
    // w is (D, 4): wv[dvec*4 + c] = the 4 taps of channel 4*dvec+c.
    const float4* wv = (const float4*)w;
    const float4 q0 = wv[dvec * 4 + 0];
    const float4 q1 = wv[dvec * 4 + 1];
    const float4 q2 = wv[dvec * 4 + 2];
    const float4 q3 = wv[dvec * 4 + 3];
    // tapK.component_c = w[4*dvec + c][K]
    const float4 tap0 = make_float4(q0.x, q1.x, q2.x, q3.x);
    const float4 tap1 = make_float4(q0.y, q1.y, q2.y, q3.y);
    const float4 tap2 = make_float4(q0.z, q1.z, q2.z, q3.z);
    const float4 tap3 = make_float4(q0.w, q1.w, q2.w, q3.w);

    // ---- consume rows with pipelined waits: async loads from one wave
    //      complete in order, so row r is resident once ASYNCcnt <= kTT-1-r.
    //      Each thread reads only LDS it wrote itself -> no barrier. ----
#pragma unroll
    for (int r = 0; r < kTT; ++r) {
        asm volatile("s_wait_asynccnt %0" :: "i"(kTT - 1 - r) : "memory");
        const float4 xc = tile[r * kBLK + tid];   // x[b, t0+r, dvec]
        v4f o;
        o.x = fmaf(tap0.x, xm3.x, fmaf(tap1.x, xm2.x, fmaf(tap2.x, xm1.x, tap3.x * xc.x)));
        o.y = fmaf(tap0.y, xm3.y, fmaf(tap1.y, xm2.y, fmaf(tap2.y, xm1.y, tap3.y * xc.y)));
        o.z = fmaf(tap0.z, xm3.z, fmaf(tap1.z, xm2.z, fmaf(tap2.z, xm1.z, tap3.z * xc.z)));
        o.w = fmaf(tap0.w, xm3.w, fmaf(tap1.w, xm2.w, fmaf(tap2.w, xm1.w, tap3.w * xc.w)));
        // y is write-once/never-read: non-temporal store keeps the 192 MB L2
        // free for the x stream (halo re-reads hit L2).
        __builtin_nontemporal_store(o, (v4f*)(yv + col + (size_t)(t0 + r) * kDV));
        xm3 = xm2;
        xm2 = xm1;
        xm1 = xc;
    }
}

extern "C" void kernel_launch(void* const* d_in, const int* in_sizes, int n_in,
                              void* d_out, int out_size, void* d_ws, size_t ws_size,
                              hipStream_t stream) {
    const float* x = (const float*)d_in[0];   // (4, 4096, 2048) f32
    const float* w = (const float*)d_in[1];   // (2048, 4) f32
    float*       y = (float*)d_out;           // (4, 4096, 2048) f32

    dim3 grid(kDV / kBLK, kT / kTT, kB);      // (4, 256, 4)
    dim3 block(kBLK);                         // 128 threads = 4 waves
    dwconv1d_async_kernel<<<grid, block, 0, stream>>>(x, w, y);
}